// DeformableDETREncoderLayer_23811298689648
// MI455X (gfx1250) — compile-verified
//
#include <hip/hip_runtime.h>

typedef float v2f __attribute__((ext_vector_type(2)));
typedef float v8f __attribute__((ext_vector_type(8)));

#define N_TOK 21760
#define M_ROWS 43520   // B * N_TOK

enum { F_RELU = 1, F_RESID = 2, F_MASK = 4 };

__device__ inline float4 add4(float4 a, float4 b) {
  a.x += b.x; a.y += b.y; a.z += b.z; a.w += b.w; return a;
}

// ---------------------------------------------------------------------------
// Generic K=256 GEMM: C[M,NOUT] = epilogue(A (+A2) @ W + bias)
// block = 256 threads (8 waves); block tile = 16 rows x 128 cols.
// Wave w computes the 16x16 tile at cols blockIdx.y*128 + w*16 using
// V_WMMA_F32_16X16X4_F32 (64 wmma per tile). NOUT/FLAGS are compile-time so
// W addressing folds into constant instruction offsets and the epilogue is
// branch-free.
// ---------------------------------------------------------------------------
template <int NOUT, int FLAGS, bool HAS_A2>
__global__ __launch_bounds__(256) void gemm_k256(
    const float* __restrict__ A, const float* __restrict__ A2,
    const float* __restrict__ W, const float* __restrict__ bias,
    const float* __restrict__ resid, const unsigned char* __restrict__ mask,
    float* __restrict__ C)
{
  __shared__ float As[16 * 260];     // stride 260: 16B-aligned, bank stride 4
  const int m0 = blockIdx.x * 16;
  const int t  = threadIdx.x;

  // Stage A tile (16 x 256) into LDS, optionally fusing A2 add (query=src+pos)
  {
    const int row = t >> 4, chunk = t & 15;
    const float4* g = (const float4*)(A + (size_t)(m0 + row) * 256 + chunk * 16);
    float4 v0 = g[0], v1 = g[1], v2 = g[2], v3 = g[3];
    if (HAS_A2) {
      const float4* g2 = (const float4*)(A2 + (size_t)(m0 + row) * 256 + chunk * 16);
      v0 = add4(v0, g2[0]); v1 = add4(v1, g2[1]);
      v2 = add4(v2, g2[2]); v3 = add4(v3, g2[3]);
    }
    float4* s = (float4*)&As[row * 260 + chunk * 16];
    s[0] = v0; s[1] = v1; s[2] = v2; s[3] = v3;
  }
  __syncthreads();

  const int wv    = t >> 5;
  const int lane  = t & 31;
  const int col   = lane & 15;
  const int koff  = (lane >> 4) << 1;   // 0 (lanes 0-15) or 2 (lanes 16-31)
  const int mhalf = (lane >> 4) << 3;   // C rows 0..7 vs 8..15
  const int n0    = blockIdx.y * 128 + wv * 16;
  const int gcol  = n0 + col;

  v8f c;
  {
    const float bv = bias[gcol];
#pragma unroll
    for (int r = 0; r < 8; ++r) c[r] = bv;
  }

  const float* arow = &As[col * 260 + koff];
  const float* wp   = W + (size_t)koff * NOUT + gcol;
#pragma unroll 16
  for (int k0 = 0; k0 < 256; k0 += 4) {
    v2f a = *(const v2f*)(arow + k0);
    v2f b;
    b.x = wp[k0 * NOUT];
    b.y = wp[(k0 + 1) * NOUT];
    c = __builtin_amdgcn_wmma_f32_16x16x4_f32(false, a, false, b, (short)0, c,
                                              false, false);
  }

  const bool zero = (FLAGS & F_MASK) ? (mask[m0 + mhalf] != 0) : false;
  (void)zero; // mask is per-row-group? no: per row; handled below per r
#pragma unroll
  for (int r = 0; r < 8; ++r) {
    const int grow = m0 + r + mhalf;
    float v = c[r];
    if (FLAGS & F_RESID) v += resid[(size_t)grow * NOUT + gcol];
    if (FLAGS & F_RELU)  v = fmaxf(v, 0.0f);
    if (FLAGS & F_MASK)  v = (mask[grow] != 0) ? 0.0f : v;
    C[(size_t)grow * NOUT + gcol] = v;
  }
}

// ---------------------------------------------------------------------------
// Deformable attention sampling: one wave per (b, query, head); lane = channel.
// Softmax over the 16 (level,point) logits is computed per-lane (uniform).
// ---------------------------------------------------------------------------
__device__ inline float corner_g(const float* __restrict__ vbase, int xi, int yi,
                                 int Wl, int Hl, float w, int ch)
{
  const bool valid = (xi >= 0) & (xi < Wl) & (yi >= 0) & (yi < Hl);
  const int xc = min(max(xi, 0), Wl - 1);
  const int yc = min(max(yi, 0), Hl - 1);
  const float v = vbase[(size_t)(yc * Wl + xc) * 256 + ch];
  return valid ? v * w : 0.0f;
}

__global__ __launch_bounds__(256) void msda_sample(
    const float* __restrict__ value, const float* __restrict__ off,
    const float* __restrict__ awl,   const float* __restrict__ ref,
    float* __restrict__ out)
{
  const int wid  = blockIdx.x * 8 + (threadIdx.x >> 5);   // over M_ROWS * HEADS
  const int lane = threadIdx.x & 31;
  const int h    = wid & 7;
  const int row  = wid >> 3;             // b*N + q
  const int b    = row / N_TOK;
  const int ch   = h * 32 + lane;

  // softmax over 16 logits of this head
  float logit[16];
  float mx = -1e30f;
#pragma unroll
  for (int j = 0; j < 16; ++j) {
    logit[j] = awl[(size_t)row * 128 + h * 16 + j];
    mx = fmaxf(mx, logit[j]);
  }
  float se = 0.0f;
#pragma unroll
  for (int j = 0; j < 16; ++j) { logit[j] = expf(logit[j] - mx); se += logit[j]; }
  const float inv = 1.0f / se;

  const int HW[4]    = {128, 64, 32, 16};
  const int START[4] = {0, 16384, 20480, 21504};

  float acc = 0.0f;
#pragma unroll
  for (int l = 0; l < 4; ++l) {
    const int   Wl = HW[l], Hl = HW[l];
    const float fW = (float)Wl, fH = (float)Hl;
    const float rx = ref[((size_t)row * 4 + l) * 2 + 0];
    const float ry = ref[((size_t)row * 4 + l) * 2 + 1];
    const float* vbase = value + (size_t)(b * N_TOK + START[l]) * 256;
#pragma unroll
    for (int p = 0; p < 4; ++p) {
      const size_t oi = (size_t)row * 256 + (size_t)((h * 4 + l) * 4 + p) * 2;
      // loc = ref + off/[W,H]; l{x,y} = loc*{W,H} - 0.5  =>  ref*{W,H}+off-0.5
      const float lx = rx * fW + off[oi + 0] - 0.5f;
      const float ly = ry * fH + off[oi + 1] - 0.5f;
      const float x0f = floorf(lx), y0f = floorf(ly);
      const float wx1 = lx - x0f,  wy1 = ly - y0f;
      const float wx0 = 1.0f - wx1, wy0 = 1.0f - wy1;
      const int x0 = (int)x0f, y0 = (int)y0f;
      float g = 0.0f;
      g += corner_g(vbase, x0,     y0,     Wl, Hl, wx0 * wy0, ch);
      g += corner_g(vbase, x0 + 1, y0,     Wl, Hl, wx1 * wy0, ch);
      g += corner_g(vbase, x0,     y0 + 1, Wl, Hl, wx0 * wy1, ch);
      g += corner_g(vbase, x0 + 1, y0 + 1, Wl, Hl, wx1 * wy1, ch);
      acc += (logit[l * 4 + p] * inv) * g;
    }
  }
  out[(size_t)row * 256 + ch] = acc;
}

// ---------------------------------------------------------------------------
// In-place LayerNorm over 256 cols; one wave per row.
// ---------------------------------------------------------------------------
__global__ __launch_bounds__(256) void ln_k(
    float* __restrict__ X, const float* __restrict__ scale,
    const float* __restrict__ bias)
{
  const int row  = blockIdx.x * 8 + (threadIdx.x >> 5);
  const int lane = threadIdx.x & 31;
  float v[8];
  float s = 0.0f;
#pragma unroll
  for (int i = 0; i < 8; ++i) { v[i] = X[(size_t)row * 256 + lane + i * 32]; s += v[i]; }
#pragma unroll
  for (int o = 16; o > 0; o >>= 1) s += __shfl_xor(s, o, 32);
  const float mean = s * (1.0f / 256.0f);
  float q = 0.0f;
#pragma unroll
  for (int i = 0; i < 8; ++i) { const float d = v[i] - mean; q += d * d; }
#pragma unroll
  for (int o = 16; o > 0; o >>= 1) q += __shfl_xor(q, o, 32);
  const float rstd = rsqrtf(q * (1.0f / 256.0f) + 1e-6f);
#pragma unroll
  for (int i = 0; i < 8; ++i) {
    const int cidx = lane + i * 32;
    X[(size_t)row * 256 + cidx] = (v[i] - mean) * rstd * scale[cidx] + bias[cidx];
  }
}

// ---------------------------------------------------------------------------
// Fused FFN: Y = X + relu(X@W1 + b1)@W2 + b2, never materializing H in HBM.
// One block per 16-row stripe; H built 512 cols at a time in LDS while the
// two 16x16 y-accumulator tiles per wave persist across both K halves.
// ---------------------------------------------------------------------------
__global__ __launch_bounds__(256) void ffn_fused(
    const float* __restrict__ X,  const float* __restrict__ W1,
    const float* __restrict__ b1, const float* __restrict__ W2,
    const float* __restrict__ b2, float* __restrict__ Y)
{
  __shared__ float Xs[16 * 260];
  __shared__ float Hs[16 * 516];
  const int m0 = blockIdx.x * 16;
  const int t  = threadIdx.x;
  {
    const int row = t >> 4, chunk = t & 15;
    const float4* g = (const float4*)(X + (size_t)(m0 + row) * 256 + chunk * 16);
    float4* s = (float4*)&Xs[row * 260 + chunk * 16];
    s[0] = g[0]; s[1] = g[1]; s[2] = g[2]; s[3] = g[3];
  }
  __syncthreads();

  const int wv    = t >> 5;
  const int lane  = t & 31;
  const int col   = lane & 15;
  const int koff  = (lane >> 4) << 1;
  const int mhalf = (lane >> 4) << 3;

  v8f yc[2];
#pragma unroll
  for (int nt = 0; nt < 2; ++nt) {
    const float bv = b2[wv * 32 + nt * 16 + col];
#pragma unroll
    for (int r = 0; r < 8; ++r) yc[nt][r] = bv;
  }

  for (int half = 0; half < 2; ++half) {
    // Stage 1: H[:, half*512 : half*512+512] = relu(X @ W1 + b1) into LDS
    for (int nt = 0; nt < 4; ++nt) {
      const int n0h   = wv * 64 + nt * 16;        // local col base in [0,512)
      const int gcol1 = half * 512 + n0h + col;   // global col in [0,1024)
      v8f c;
      {
        const float bv = b1[gcol1];
#pragma unroll
        for (int r = 0; r < 8; ++r) c[r] = bv;
      }
      const float* arow = &Xs[col * 260 + koff];
      const float* wp   = W1 + (size_t)koff * 1024 + gcol1;
#pragma unroll 16
      for (int k0 = 0; k0 < 256; k0 += 4) {
        v2f a = *(const v2f*)(arow + k0);
        v2f b;
        b.x = wp[k0 * 1024];
        b.y = wp[(k0 + 1) * 1024];
        c = __builtin_amdgcn_wmma_f32_16x16x4_f32(false, a, false, b, (short)0,
                                                  c, false, false);
      }
#pragma unroll
      for (int r = 0; r < 8; ++r)
        Hs[(r + mhalf) * 516 + n0h + col] = fmaxf(c[r], 0.0f);
    }
    __syncthreads();

    // Stage 2 (partial): yc += H_half @ W2[half*512 : half*512+512, :]
#pragma unroll
    for (int nt = 0; nt < 2; ++nt) {
      const int gcol = wv * 32 + nt * 16 + col;
      const float* arow = &Hs[col * 516 + koff];
      const float* wp   = W2 + (size_t)(half * 512 + koff) * 256 + gcol;
      v8f c = yc[nt];
#pragma unroll 16
      for (int k0 = 0; k0 < 512; k0 += 4) {
        v2f a = *(const v2f*)(arow + k0);
        v2f b;
        b.x = wp[k0 * 256];
        b.y = wp[(k0 + 1) * 256];
        c = __builtin_amdgcn_wmma_f32_16x16x4_f32(false, a, false, b, (short)0,
                                                  c, false, false);
      }
      yc[nt] = c;
    }
    __syncthreads();
  }

  // Epilogue: Y = yc + X (residual), pre-LN2
#pragma unroll
  for (int nt = 0; nt < 2; ++nt) {
    const int ncol = wv * 32 + nt * 16 + col;
#pragma unroll
    for (int r = 0; r < 8; ++r) {
      const int mrow = r + mhalf;
      Y[(size_t)(m0 + mrow) * 256 + ncol] = yc[nt][r] + Xs[mrow * 260 + ncol];
    }
  }
}

// ---------------------------------------------------------------------------
extern "C" void kernel_launch(void* const* d_in, const int* in_sizes, int n_in,
                              void* d_out, int out_size, void* d_ws, size_t ws_size,
                              hipStream_t stream) {
  const float* src   = (const float*)d_in[0];
  const float* pos   = (const float*)d_in[1];
  const float* ref   = (const float*)d_in[2];
  const unsigned char* pad = (const unsigned char*)d_in[3];
  const float* Wv    = (const float*)d_in[4];
  const float* bv    = (const float*)d_in[5];
  const float* Woff  = (const float*)d_in[6];
  const float* boff  = (const float*)d_in[7];
  const float* Wattn = (const float*)d_in[8];
  const float* battn = (const float*)d_in[9];
  const float* Wout  = (const float*)d_in[10];
  const float* bout  = (const float*)d_in[11];
  const float* ln1s  = (const float*)d_in[12];
  const float* ln1b  = (const float*)d_in[13];
  const float* W1    = (const float*)d_in[14];
  const float* b1    = (const float*)d_in[15];
  const float* W2    = (const float*)d_in[16];
  const float* b2    = (const float*)d_in[17];
  const float* ln2s  = (const float*)d_in[18];
  const float* ln2b  = (const float*)d_in[19];
  float* out = (float*)d_out;

  const size_t M = M_ROWS;
  float* bval  = (float*)d_ws;            // [M,256] value
  float* btmp  = bval  + M * 256;         // [M,256] off -> (src+attn) -> x
  float* bawl  = btmp  + M * 256;         // [M,128] attn logits
  float* bsamp = bawl  + M * 128;         // [M,256] sampled

  const dim3 blk(256);

  // value = where(pad, 0, src @ Wv + bv)
  gemm_k256<256, F_MASK, false><<<dim3(M / 16, 2), blk, 0, stream>>>(
      src, nullptr, Wv, bv, nullptr, pad, bval);
  // off = (src+pos) @ Woff + boff
  gemm_k256<256, 0, true><<<dim3(M / 16, 2), blk, 0, stream>>>(
      src, pos, Woff, boff, nullptr, nullptr, btmp);
  // attn logits = (src+pos) @ Wattn + battn
  gemm_k256<128, 0, true><<<dim3(M / 16, 1), blk, 0, stream>>>(
      src, pos, Wattn, battn, nullptr, nullptr, bawl);
  // deformable sampling (softmax fused): bsamp[M,256]
  msda_sample<<<dim3(M), blk, 0, stream>>>(bval, btmp, bawl, ref, bsamp);
  // tmp = src + bsamp @ Wout + bout
  gemm_k256<256, F_RESID, false><<<dim3(M / 16, 2), blk, 0, stream>>>(
      bsamp, nullptr, Wout, bout, src, nullptr, btmp);
  // x = LN1(tmp) in place
  ln_k<<<dim3(M / 8), blk, 0, stream>>>(btmp, ln1s, ln1b);
  // out = x + relu(x@W1+b1)@W2 + b2
  ffn_fused<<<dim3(M / 16), blk, 0, stream>>>(btmp, W1, b1, W2, b2, out);
  // out = LN2(out) in place
  ln_k<<<dim3(M / 8), blk, 0, stream>>>(out, ln2s, ln2b);
}